// TripletLoss_81441169867288
// MI455X (gfx1250) — compile-verified
//
#include <hip/hip_runtime.h>

#define BATCH 512
#define FEAT  256
#define MARGIN 0.2f

typedef float v2f __attribute__((ext_vector_type(2)));
typedef float v8f __attribute__((ext_vector_type(8)));

// ---------------------------------------------------------------------------
// Kernel 0: per-row squared norms  sq[i] = sum_k X[i][k]^2
// ---------------------------------------------------------------------------
__global__ __launch_bounds__(128) void sq_kernel(const float* __restrict__ x,
                                                 float* __restrict__ sq) {
    const int row = blockIdx.x * blockDim.x + threadIdx.x;
    if (row >= BATCH) return;
    const float4* r = reinterpret_cast<const float4*>(x + row * FEAT);
    float s = 0.0f;
#pragma unroll 4
    for (int c = 0; c < FEAT / 4; ++c) {
        float4 v = r[c];
        s += v.x * v.x + v.y * v.y + v.z * v.z + v.w * v.w;
    }
    sq[row] = s;
}

// ---------------------------------------------------------------------------
// Kernel 1: pdist tile via V_WMMA_F32_16X16X4_F32 (exact f32 matrix path).
// One wave (32 lanes) per 16x16 tile of gram = X X^T.
//   A layout (16x4 f32): lane<16 -> A[M=lane][K=0..1], lane>=16 -> A[..][K=2..3]
//   B layout mirrors A with M<->N; since B = X^T tile, loads look like A's.
//   C/D: VGPR r = row 8*(lane>=16)+r, col lane%16.
// pdist[i][j] = sq[i] + sq[j] - 2*gram[i][j]
// ---------------------------------------------------------------------------
__global__ __launch_bounds__(32) void pdist_wmma_kernel(const float* __restrict__ x,
                                                        const float* __restrict__ sq,
                                                        float* __restrict__ pdist) {
    const int lane = threadIdx.x;      // 0..31, EXEC all-ones (WMMA requirement)
    const int m    = lane & 15;
    const int kq   = lane >> 4;        // 0: K pair {0,1}, 1: K pair {2,3}
    const int i0   = blockIdx.y * 16;
    const int j0   = blockIdx.x * 16;

    const float* arow = x + (i0 + m) * FEAT + 2 * kq;
    const float* brow = x + (j0 + m) * FEAT + 2 * kq;

    v8f acc = {};
#pragma unroll 8
    for (int k = 0; k < FEAT; k += 4) {
        v2f a = *reinterpret_cast<const v2f*>(arow + k);   // 8B aligned b64 load
        v2f b = *reinterpret_cast<const v2f*>(brow + k);
        // 8 args: (neg_a, A, neg_b, B, c_mod, C, reuse_a, reuse_b)
        acc = __builtin_amdgcn_wmma_f32_16x16x4_f32(
            false, a, false, b, (short)0, acc, false, false);
    }

    const int   gj  = j0 + m;
    const float sqj = sq[gj];
#pragma unroll
    for (int r = 0; r < 8; ++r) {
        const int gi = i0 + 8 * kq + r;
        pdist[gi * BATCH + gj] = sq[gi] + sqj - 2.0f * acc[r];
    }
}

// ---------------------------------------------------------------------------
// Kernel 2: per-anchor triplet partial sums.
// For anchor i: P = {j : j!=i, y_j==y_i}, N = {k : y_k!=y_i}  (j!=k, i!=k implied)
//   pHinge[i] = sum_{j in P} sum_{k in N} max(d_ij - d_ik + m, 0)
//   pCnt[i]   = |P| * |N|
// Deterministic shared-memory tree reductions (no float atomics).
// ---------------------------------------------------------------------------
__global__ __launch_bounds__(256) void triplet_kernel(const float* __restrict__ pdist,
                                                      const int* __restrict__ y,
                                                      float* __restrict__ pHinge,
                                                      float* __restrict__ pCnt) {
    __shared__ float s_d[BATCH];
    __shared__ int   s_y[BATCH];
    __shared__ float s_red[256];

    const int i = blockIdx.x;
    const int t = threadIdx.x;

    s_d[t]       = pdist[i * BATCH + t];
    s_d[t + 256] = pdist[i * BATCH + t + 256];
    s_y[t]       = y[t];
    s_y[t + 256] = y[t + 256];
    __syncthreads();

    const int yi = s_y[i];

    // negatives handled by this thread's two k-slots
    float negLocal = (float)((s_y[t] != yi) + (s_y[t + 256] != yi));

    float posLocal = 0.0f;
    float hinge    = 0.0f;
#pragma unroll
    for (int jj = 0; jj < 2; ++jj) {
        const int j = t + jj * 256;
        if (j != i && s_y[j] == yi) {
            posLocal += 1.0f;
            const float dij = s_d[j];
            for (int k = 0; k < BATCH; ++k) {
                if (s_y[k] != yi) {
                    hinge += fmaxf(dij - s_d[k] + MARGIN, 0.0f);
                }
            }
        }
    }

    // --- reduce hinge ---
    s_red[t] = hinge;
    __syncthreads();
    for (int s = 128; s > 0; s >>= 1) {
        if (t < s) s_red[t] += s_red[t + s];
        __syncthreads();
    }
    const float hSum = s_red[0];
    __syncthreads();

    // --- reduce positive count ---
    s_red[t] = posLocal;
    __syncthreads();
    for (int s = 128; s > 0; s >>= 1) {
        if (t < s) s_red[t] += s_red[t + s];
        __syncthreads();
    }
    const float pSum = s_red[0];
    __syncthreads();

    // --- reduce negative count ---
    s_red[t] = negLocal;
    __syncthreads();
    for (int s = 128; s > 0; s >>= 1) {
        if (t < s) s_red[t] += s_red[t + s];
        __syncthreads();
    }

    if (t == 0) {
        pHinge[i] = hSum;
        pCnt[i]   = pSum * s_red[0];   // |P| * |N|  (exact in f32, < 2^24)
    }
}

// ---------------------------------------------------------------------------
// Kernel 3: final deterministic reduction, out = sum(hinge) / sum(count)
// ---------------------------------------------------------------------------
__global__ __launch_bounds__(256) void finalize_kernel(const float* __restrict__ pHinge,
                                                       const float* __restrict__ pCnt,
                                                       float* __restrict__ out) {
    __shared__ float sh[256];
    __shared__ float sc[256];
    const int t = threadIdx.x;
    sh[t] = pHinge[t] + pHinge[t + 256];
    sc[t] = pCnt[t]   + pCnt[t + 256];
    __syncthreads();
    for (int s = 128; s > 0; s >>= 1) {
        if (t < s) { sh[t] += sh[t + s]; sc[t] += sc[t + s]; }
        __syncthreads();
    }
    if (t == 0) out[0] = sh[0] / sc[0];
}

// ---------------------------------------------------------------------------
extern "C" void kernel_launch(void* const* d_in, const int* in_sizes, int n_in,
                              void* d_out, int out_size, void* d_ws, size_t ws_size,
                              hipStream_t stream) {
    (void)in_sizes; (void)n_in; (void)out_size; (void)ws_size;

    const float* x = (const float*)d_in[0];
    const int*   y = (const int*)d_in[1];
    float*       out = (float*)d_out;

    float* ws     = (float*)d_ws;
    float* sq     = ws;                         // 512 floats
    float* pdist  = sq + BATCH;                 // 512*512 floats (1 MB)
    float* pHinge = pdist + BATCH * BATCH;      // 512 floats
    float* pCnt   = pHinge + BATCH;             // 512 floats

    sq_kernel<<<BATCH / 128, 128, 0, stream>>>(x, sq);

    dim3 grid(BATCH / 16, BATCH / 16);          // 32x32 tiles, 1 wave each
    pdist_wmma_kernel<<<grid, 32, 0, stream>>>(x, sq, pdist);

    triplet_kernel<<<BATCH, 256, 0, stream>>>(pdist, y, pHinge, pCnt);

    finalize_kernel<<<1, 256, 0, stream>>>(pHinge, pCnt, out);
}